// E3Hamiltonian_62019327754825
// MI455X (gfx1250) — compile-verified
//
#include <hip/hip_runtime.h>
#include <stdint.h>

// ---------------------------------------------------------------------------
// E3 Hamiltonian block expansion. Memory-bound: 600k edges x 282 f32 in/out
// = ~1.35 GB @ 23.3 TB/s => ~58 us floor. f32 end-to-end, math on
// V_WMMA_F32_16X16X4_F32, staging via global_load_async_to_lds_b64.
// ---------------------------------------------------------------------------

typedef float v2f __attribute__((ext_vector_type(2)));
typedef float v8f __attribute__((ext_vector_type(8)));

static constexpr int RED          = 282;
static constexpr int N_EDGE       = 600000;
static constexpr int EDGES_PER_WG = 32;       // 2 waves x 16 edges
static constexpr int THREADS      = 64;
static constexpr int FSTRIDE      = 284;      // padded row stride: bank-conflict-free
static constexpr int LFEAT_FLOATS = EDGES_PER_WG * FSTRIDE + 8;   // + zeroed tail
static constexpr int BPAD_FLOATS  = 1536;     // 64+128+192+256+896

// Build zero-padded B table in LDS:  Bpad[k][col] = cg[col*M + k], 0 outside.
// Padding K to KP (x4) and N to W (x16) lets ALL fragment gathers be
// unconditional: A-values hitting k>=M multiply exact zeros.
template <int M, int KP, int W>
__device__ __forceinline__ void build_bpad(float* __restrict__ bdst,
                                           const float* __restrict__ cg, int t) {
  for (int i = t; i < KP * W; i += THREADS) {
    const int k = i / W, col = i % W;
    bdst[i] = (k < M && col < M) ? cg[col * M + k] : 0.0f;
  }
}

// One orbital-pair block as NPAIR tiny GEMMs on the f32 WMMA pipe.
// A (16x4): lane%16 = edge row, regs hold K = 2*(lane/16)+{0,1}
// B (4x16): lane%16 = col,      regs hold K = 2*(lane/16)+{0,1}
// D (16x16): lane%16 = col, VGPR v -> edge row v + 8*(lane/16)
template <int OFF, int NPAIR, int M>
__device__ __forceinline__ void block_wmma(const float* __restrict__ lfeatp,
                                           const float* __restrict__ bpad,
                                           float* __restrict__ out,
                                           int gbase, int wave, int lane) {
  constexpr int KS = (M + 3) / 4;
  constexpr int NT = (M + 15) / 16;
  constexpr int W  = NT * 16;
  const int half = lane >> 4;
  const int l16  = lane & 15;
  const float* arow = lfeatp + (wave * 16 + l16) * FSTRIDE + OFF;

#pragma unroll
  for (int nt = 0; nt < NT; ++nt) {
    // B fragments for this N-tile: unconditional LDS reads, reused by all pairs.
    v2f bf[KS];
#pragma unroll
    for (int ks = 0; ks < KS; ++ks) {
      const int k = ks * 4 + half * 2;
      const float* bp = bpad + k * W + nt * 16 + l16;
      v2f b; b.x = bp[0]; b.y = bp[W];
      bf[ks] = b;
    }

#pragma unroll
    for (int p0 = 0; p0 < NPAIR; p0 += 2) {
      // Prefetch A fragments for two pairs as one ds_load clause.
      v2f a0[KS], a1[KS];
#pragma unroll
      for (int ks = 0; ks < KS; ++ks) {
        const int k = ks * 4 + half * 2;
        a0[ks].x = arow[p0 * M + k];
        a0[ks].y = arow[p0 * M + k + 1];
        if (p0 + 1 < NPAIR) {                 // folds at compile time (unrolled)
          a1[ks].x = arow[(p0 + 1) * M + k];
          a1[ks].y = arow[(p0 + 1) * M + k + 1];
        }
      }
      // Two independent accumulator chains -> WMMA co-execution / ILP.
      v8f acc0 = {0,0,0,0,0,0,0,0};
      v8f acc1 = {0,0,0,0,0,0,0,0};
#pragma unroll
      for (int ks = 0; ks < KS; ++ks) {
        acc0 = __builtin_amdgcn_wmma_f32_16x16x4_f32(
            false, a0[ks], false, bf[ks], (short)0, acc0, false, false);
        if (p0 + 1 < NPAIR)
          acc1 = __builtin_amdgcn_wmma_f32_16x16x4_f32(
              false, a1[ks], false, bf[ks], (short)0, acc1, false, false);
      }
      const int colg = nt * 16 + l16;
      if (colg < M) {
#pragma unroll
        for (int v = 0; v < 8; ++v) {
          const size_t e = (size_t)(gbase + wave * 16 + v + half * 8);
          out[e * RED + OFF + p0 * M + colg] = acc0[v];
          if (p0 + 1 < NPAIR)
            out[e * RED + OFF + (p0 + 1) * M + colg] = acc1[v];
        }
      }
    }
  }
}

extern "C" __global__ void __launch_bounds__(THREADS)
e3ham_wmma_kernel(const float* __restrict__ feat,
                  const float* __restrict__ cg_ss,
                  const float* __restrict__ cg_sp,
                  const float* __restrict__ cg_sd,
                  const float* __restrict__ cg_pp,
                  const float* __restrict__ cg_pd,
                  const float* __restrict__ cg_dd,
                  float* __restrict__ out) {
  __shared__ float lfeatp[LFEAT_FLOATS];   // 32 x 284 f32 + tail pad
  __shared__ float bpad[BPAD_FLOATS];

  const int t = threadIdx.x;

  // Zero-pad columns 282/283 of each row + tail (so A overreads are finite 0s).
  if (t < EDGES_PER_WG) {
    lfeatp[t * FSTRIDE + 282] = 0.0f;
    lfeatp[t * FSTRIDE + 283] = 0.0f;
  }
  if (t < 8) lfeatp[EDGES_PER_WG * FSTRIDE + t] = 0.0f;

  // Zero-padded B tables (once per workgroup).
  build_bpad< 3,  4, 16>(bpad +   0, cg_sp, t);
  build_bpad< 5,  8, 16>(bpad +  64, cg_sd, t);
  build_bpad< 9, 12, 16>(bpad + 192, cg_pp, t);
  build_bpad<15, 16, 16>(bpad + 384, cg_pd, t);
  build_bpad<25, 28, 32>(bpad + 640, cg_dd, t);

  // Stage 32 edge rows (282 f32 each) into the stride-284 LDS tile using the
  // CDNA5 async direct-to-LDS path (8B units keep both sides aligned).
  {
    const float* gsrc = feat + (size_t)blockIdx.x * (EDGES_PER_WG * RED);
    constexpr int N2 = EDGES_PER_WG * (RED / 2);   // 4512 float2 units
    for (int i = t; i < N2; i += THREADS) {
      const int row = i / 141;
      const int w2  = i - row * 141;
      const unsigned lds_byte =
          (unsigned)(uintptr_t)&lfeatp[row * FSTRIDE + w2 * 2];
      const unsigned long long gaddr =
          (unsigned long long)(uintptr_t)(gsrc + i * 2);
      asm volatile("global_load_async_to_lds_b64 %0, %1, off"
                   :: "v"(lds_byte), "v"(gaddr) : "memory");
    }
    asm volatile("s_wait_asynccnt 0" ::: "memory");
  }
  __syncthreads();

  const int wave  = t >> 5;
  const int lane  = t & 31;
  const int gbase = blockIdx.x * EDGES_PER_WG;   // 600000 % 32 == 0: no guards

  // ss block (m == 1): uniform scalar scale.
  {
    const float s = cg_ss[0];
    for (int i = t; i < EDGES_PER_WG * 6; i += THREADS) {
      const int e = i / 6, p = i - e * 6;
      out[(size_t)(gbase + e) * RED + p] = s * lfeatp[e * FSTRIDE + p];
    }
  }

  //            OFF  P   M          Bpad offset
  block_wmma<  6, 9,  3>(lfeatp, bpad +   0, out, gbase, wave, lane);  // sp
  block_wmma< 33, 6,  5>(lfeatp, bpad +  64, out, gbase, wave, lane);  // sd
  block_wmma< 63, 6,  9>(lfeatp, bpad + 192, out, gbase, wave, lane);  // pp
  block_wmma<117, 6, 15>(lfeatp, bpad + 384, out, gbase, wave, lane);  // pd
  block_wmma<207, 3, 25>(lfeatp, bpad + 640, out, gbase, wave, lane);  // dd
}

extern "C" void kernel_launch(void* const* d_in, const int* in_sizes, int n_in,
                              void* d_out, int out_size, void* d_ws, size_t ws_size,
                              hipStream_t stream) {
  (void)in_sizes; (void)n_in; (void)out_size; (void)d_ws; (void)ws_size;
  const float* feat  = (const float*)d_in[0];
  const float* cg_ss = (const float*)d_in[1];
  const float* cg_sp = (const float*)d_in[2];
  const float* cg_sd = (const float*)d_in[3];
  const float* cg_pp = (const float*)d_in[4];
  const float* cg_pd = (const float*)d_in[5];
  const float* cg_dd = (const float*)d_in[6];
  float* out = (float*)d_out;

  const int nblocks = N_EDGE / EDGES_PER_WG;   // 18750, exact
  e3ham_wmma_kernel<<<dim3(nblocks), dim3(THREADS), 0, stream>>>(
      feat, cg_ss, cg_sp, cg_sd, cg_pp, cg_pd, cg_dd, out);
}